// MambaSSM_429496729845
// MI455X (gfx1250) — compile-verified
//
#include <hip/hip_runtime.h>
#include <hip/hip_bf16.h>

// ---------------------------------------------------------------------------
// Mamba-UNet hybrid for gfx1250 (MI455X).
// All matmuls (convs via im2col + projections) use V_WMMA_F32_16X16X4_F32
// (full f32 precision matrix pipe). The sequential SSM scan runs one wave32
// per (batch, d_inner) recurrence with 2 states per lane in registers.
// ---------------------------------------------------------------------------

#define D_MODEL 256
#define D_INNER 512
#define D_STATE 64
#define DT_RANK 16
#define BSZ     2
#define HH      64
#define WW      64
#define HWSZ    (HH * WW)     // 4096
#define LSEQ    HWSZ          // 4096
#define MROWS   (BSZ * LSEQ)  // 8192

typedef __attribute__((ext_vector_type(2))) float v2f;
typedef __attribute__((ext_vector_type(8))) float v8f;

// ======================= WMMA f32 GEMM =====================================
// C(MxN) = act( A(MxKp) * Bt(KpxNp) + bias ),  Bt is pre-packed K-major,
// zero padded to Kp (mult of 4) x Np (mult of 16) so the hot loop is
// branch-free.  A can be row-major (a_hw==0) or NCHW with HW=a_hw (K=C).
// Output can be row-major (c_hw==0) or scattered to NCHW with HW=c_hw.
// act: 0=none 1=relu 2=softplus
__global__ __launch_bounds__(256) void ms_gemm_wmma_f32(
    const float* __restrict__ A, int lda, int a_hw,
    const float* __restrict__ Bt, int ldb /*==Np*/,
    const float* __restrict__ bias,
    float* __restrict__ Co, int ldc, int c_hw,
    int M, int N, int Kp, int act)
{
  const int lane = threadIdx.x & 31;
  const int wave = threadIdx.x >> 5;     // 8 waves: 2 (M) x 4 (N)
  const int mt = blockIdx.y * 2 + (wave >> 2);
  const int nt = blockIdx.x * 4 + (wave & 3);
  const int Np = ldb;
  if (mt * 16 >= M || nt * 16 >= Np) return;   // wave-uniform

  // A 16x4 layout: lanes 0-15 -> M=lane, K=0,1 ; lanes 16-31 -> M=lane-16, K=2,3
  // B 4x16 layout mirrored with N on lanes.
  const int mrow = mt * 16 + (lane & 15);
  const int ncol = nt * 16 + (lane & 15);
  const int koff = (lane >> 4) * 2;

  v8f c = {};
  const float* bp = Bt + (size_t)koff * ldb + ncol;

  if (a_hw > 0) {
    const int bidx = mrow / a_hw;
    const int tpos = mrow % a_hw;
    const float* ap = A + ((size_t)bidx * Kp + koff) * a_hw + tpos;
    for (int k = 0; k < Kp; k += 4) {
      v2f a, b;
      a.x = ap[0];   a.y = ap[a_hw];
      b.x = bp[0];   b.y = bp[ldb];
      c = __builtin_amdgcn_wmma_f32_16x16x4_f32(false, a, false, b,
                                                (short)0, c, false, false);
      ap += (size_t)4 * a_hw;
      bp += (size_t)4 * ldb;
    }
  } else {
    const float* ap = A + (size_t)mrow * lda + koff;
    for (int k = 0; k < Kp; k += 4) {
      v2f a, b;
      a.x = ap[0];   a.y = ap[1];
      b.x = bp[0];   b.y = bp[ldb];
      c = __builtin_amdgcn_wmma_f32_16x16x4_f32(false, a, false, b,
                                                (short)0, c, false, false);
      ap += 4;
      bp += (size_t)4 * ldb;
    }
  }

  const int n_out = nt * 16 + (lane & 15);
  if (n_out < N) {
    const float bv = bias ? bias[n_out] : 0.0f;
    #pragma unroll
    for (int v = 0; v < 8; ++v) {
      const int m_out = mt * 16 + (lane >> 4) * 8 + v;
      float val = c[v] + bv;
      if (act == 1)      val = fmaxf(val, 0.0f);
      else if (act == 2) val = (val > 20.0f) ? val : log1pf(expf(val));
      size_t idx;
      if (c_hw > 0) idx = ((size_t)(m_out / c_hw) * N + n_out) * c_hw + (m_out % c_hw);
      else          idx = (size_t)m_out * ldc + n_out;
      Co[idx] = val;
    }
  }
}

// ======================= packing kernels ===================================
// Bt[k*Np+n] = W^T with zero padding.  W is (N x K) row-major.
__global__ void ms_pack_proj(const float* __restrict__ W, float* __restrict__ Bt,
                             int N, int K, int Kp, int Np, long total)
{
  long i = (long)blockIdx.x * blockDim.x + threadIdx.x;
  if (i >= total) return;
  int n = (int)(i % Np);
  int k = (int)(i / Np);
  Bt[i] = (k < K && n < N) ? W[(size_t)n * K + k] : 0.0f;
}

// Conv weights (O,I,3,3) -> Bt[(c*9+kh*3+kw)][o]
__global__ void ms_pack_conv(const float* __restrict__ W, float* __restrict__ Bt,
                             int O, int I, int Kp, int Np, long total)
{
  long i = (long)blockIdx.x * blockDim.x + threadIdx.x;
  if (i >= total) return;
  int n = (int)(i % Np);
  int k = (int)(i / Np);
  float v = 0.0f;
  if (n < O && k < I * 9) {
    int c = k / 9, r = k % 9;
    v = W[(size_t)(n * I + c) * 9 + r];
  }
  Bt[i] = v;
}

// Deconv weights (I,O,3,3): equivalent conv weight w_eq[o][c][kh][kw] =
// W[c][o][2-kh][2-kw]  -> Bt[(c*9+kh*3+kw)][o]
__global__ void ms_pack_deconv(const float* __restrict__ W, float* __restrict__ Bt,
                               int O, int I, int Kp, int Np, long total)
{
  long i = (long)blockIdx.x * blockDim.x + threadIdx.x;
  if (i >= total) return;
  int n = (int)(i % Np);
  int k = (int)(i / Np);
  float v = 0.0f;
  if (n < O && k < I * 9) {
    int c = k / 9, r = k % 9, kh = r / 3, kw = r % 3;
    v = W[((size_t)(c * O + n) * 3 + (2 - kh)) * 3 + (2 - kw)];
  }
  Bt[i] = v;
}

// im2col (3x3, pad 1) from NCHW, optional fused residual add, zero-padded K.
__global__ void ms_im2col(const float* __restrict__ in, const float* __restrict__ add,
                          float* __restrict__ out, int C, int Kp, long total)
{
  long i = (long)blockIdx.x * blockDim.x + threadIdx.x;
  if (i >= total) return;
  int  k = (int)(i % Kp);
  long m = i / Kp;
  int  b   = (int)(m / HWSZ);
  int  rem = (int)(m % HWSZ);
  int  y = rem / WW, x = rem % WW;
  float v = 0.0f;
  if (k < C * 9) {
    int c = k / 9, r = k % 9, kh = r / 3, kw = r % 3;
    int iy = y + kh - 1, ix = x + kw - 1;
    if (iy >= 0 && iy < HH && ix >= 0 && ix < WW) {
      size_t idx = ((size_t)(b * C + c) * HH + iy) * WW + ix;
      v = in[idx];
      if (add) v += add[idx];
    }
  }
  out[i] = v;
}

// ======================= mamba pointwise kernels ===========================
// depthwise causal conv1d (k=4) + silu.  xz rows are 1024 wide (u = first 512).
__global__ void ms_conv1d_silu(const float* __restrict__ xz,
                               const float* __restrict__ w,
                               const float* __restrict__ bias,
                               float* __restrict__ ucv, long total)
{
  long i = (long)blockIdx.x * blockDim.x + threadIdx.x;
  if (i >= total) return;
  int  d  = (int)(i % D_INNER);
  long bt = i / D_INNER;
  int  t  = (int)(bt % LSEQ);
  int  b  = (int)(bt / LSEQ);
  float acc = bias[d];
  #pragma unroll
  for (int k = 0; k < 4; ++k) {
    int tt = t + k - 3;
    if (tt >= 0)
      acc += xz[(size_t)(b * LSEQ + tt) * 1024 + d] * w[d * 4 + k];
  }
  ucv[i] = acc / (1.0f + expf(-acc));   // silu
}

// Selective-scan: one wave32 per (b,d); lane owns states n=lane and n=lane+32.
__global__ __launch_bounds__(256) void ms_scan(const float* __restrict__ ucv,
                                               const float* __restrict__ dtb,
                                               const float* __restrict__ dbl,
                                               const float* __restrict__ A_log,
                                               float* __restrict__ ys)
{
  const int wid  = (int)((blockIdx.x * blockDim.x + threadIdx.x) >> 5);
  const int lane = threadIdx.x & 31;
  if (wid >= BSZ * D_INNER) return;          // wave-uniform
  const int b = wid / D_INNER;
  const int d = wid % D_INNER;

  const float A0 = -expf(A_log[d * D_STATE + lane]);
  const float A1 = -expf(A_log[d * D_STATE + lane + 32]);
  float h0 = 0.0f, h1 = 0.0f;

  const float* dtp = dtb + (size_t)b * LSEQ * D_INNER + d;
  const float* up  = ucv + (size_t)b * LSEQ * D_INNER + d;
  const float* Bp  = dbl + (size_t)b * LSEQ * 144 + DT_RANK;
  const float* Cp  = dbl + (size_t)b * LSEQ * 144 + DT_RANK + D_STATE;
  float*       yp  = ys  + (size_t)b * LSEQ * D_INNER + d;

  for (int t = 0; t < LSEQ; ++t) {
    const float dt = dtp[(size_t)t * D_INNER];
    const float u  = up[(size_t)t * D_INNER];
    const float du = dt * u;
    const float B0 = Bp[(size_t)t * 144 + lane];
    const float B1 = Bp[(size_t)t * 144 + lane + 32];
    const float C0 = Cp[(size_t)t * 144 + lane];
    const float C1 = Cp[(size_t)t * 144 + lane + 32];
    h0 = expf(dt * A0) * h0 + du * B0;
    h1 = expf(dt * A1) * h1 + du * B1;
    float acc = h0 * C0 + h1 * C1;
    #pragma unroll
    for (int off = 16; off >= 1; off >>= 1)
      acc += __shfl_xor(acc, off, 32);       // wave32 butterfly reduce
    if (lane == 0) yp[(size_t)t * D_INNER] = acc;
  }
}

// y = (ys + u*D) * silu(z)
__global__ void ms_gate(const float* __restrict__ ys, const float* __restrict__ ucv,
                        const float* __restrict__ xz, const float* __restrict__ Dp,
                        float* __restrict__ yb, long total)
{
  long i = (long)blockIdx.x * blockDim.x + threadIdx.x;
  if (i >= total) return;
  int  d = (int)(i % D_INNER);
  long m = i / D_INNER;
  float z = xz[(size_t)m * 1024 + D_INNER + d];
  float s = z / (1.0f + expf(-z));
  yb[i] = (ys[i] + ucv[i] * Dp[d]) * s;
}

// ======================= host side =========================================
extern "C" void kernel_launch(void* const* d_in, const int* in_sizes, int n_in,
                              void* d_out, int out_size, void* d_ws, size_t ws_size,
                              hipStream_t stream)
{
  const float* x    = (const float*)d_in[0];
  const float* c1w  = (const float*)d_in[1];
  const float* c1b  = (const float*)d_in[2];
  const float* c2w  = (const float*)d_in[3];
  const float* c2b  = (const float*)d_in[4];
  const float* c3w  = (const float*)d_in[5];
  const float* c3b  = (const float*)d_in[6];
  const float* ipw  = (const float*)d_in[7];
  const float* c1dw = (const float*)d_in[8];
  const float* c1db = (const float*)d_in[9];
  const float* xpw  = (const float*)d_in[10];
  const float* dtw  = (const float*)d_in[11];
  const float* dtbb = (const float*)d_in[12];
  const float* Alog = (const float*)d_in[13];
  const float* Dpar = (const float*)d_in[14];
  const float* opw  = (const float*)d_in[15];
  const float* d1w  = (const float*)d_in[16];
  const float* d1b  = (const float*)d_in[17];
  const float* d2w  = (const float*)d_in[18];
  const float* d2b  = (const float*)d_in[19];
  const float* d3w  = (const float*)d_in[20];
  const float* d3b  = (const float*)d_in[21];
  float* out = (float*)d_out;
  float* ws  = (float*)d_ws;

  // workspace layout (floats)
  size_t off = 0;
  auto take = [&](size_t n) { size_t o = off; off += n; return o; };
  float* im  = ws + take((size_t)MROWS * 2304);   // im2col panel (max K=2304)
  float* wp  = ws + take((size_t)2304 * 128 > 1152 * 256 ? 2304 * 128 : 1152 * 256);
  float* x1  = ws + take((size_t)BSZ * 64  * HWSZ);
  float* x2  = ws + take((size_t)BSZ * 128 * HWSZ);
  float* x3  = ws + take((size_t)BSZ * 256 * HWSZ);
  float* xz  = ws + take((size_t)MROWS * 1024);
  float* ucv = ws + take((size_t)MROWS * D_INNER);
  float* dbl = ws + take((size_t)MROWS * 144);
  float* dtb = ws + take((size_t)MROWS * D_INNER);
  float* ysb = ws + take((size_t)MROWS * D_INNER);
  float* yb  = ws + take((size_t)MROWS * D_INNER);
  float* x3s = ws + take((size_t)BSZ * 256 * HWSZ);
  float* x4  = ws + take((size_t)BSZ * 128 * HWSZ);
  float* x5  = ws + take((size_t)BSZ * 64  * HWSZ);

  const int TB = 256;
  auto blocks = [&](long n) { return (unsigned)((n + TB - 1) / TB); };

  auto gemm = [&](const float* A, int lda, int a_hw, const float* Bt, int Np,
                  const float* bias, float* C, int ldc, int c_hw,
                  int M, int N, int Kp, int act) {
    dim3 g((Np + 63) / 64, (M + 31) / 32);
    ms_gemm_wmma_f32<<<g, TB, 0, stream>>>(A, lda, a_hw, Bt, Np, bias,
                                           C, ldc, c_hw, M, N, Kp, act);
  };
  auto im2col = [&](const float* in, const float* add, int C, int Kp) {
    long tot = (long)MROWS * Kp;
    ms_im2col<<<blocks(tot), TB, 0, stream>>>(in, add, im, C, Kp, tot);
  };

  // ---- encoder convs (im2col + WMMA GEMM, NCHW output, relu) ----
  ms_pack_conv<<<blocks((long)28 * 64), TB, 0, stream>>>(c1w, wp, 64, 3, 28, 64, 28L * 64);
  im2col(x, nullptr, 3, 28);
  gemm(im, 28, 0, wp, 64, c1b, x1, 0, HWSZ, MROWS, 64, 28, 1);

  ms_pack_conv<<<blocks((long)576 * 128), TB, 0, stream>>>(c2w, wp, 128, 64, 576, 128, 576L * 128);
  im2col(x1, nullptr, 64, 576);
  gemm(im, 576, 0, wp, 128, c2b, x2, 0, HWSZ, MROWS, 128, 576, 1);

  ms_pack_conv<<<blocks((long)1152 * 256), TB, 0, stream>>>(c3w, wp, 256, 128, 1152, 256, 1152L * 256);
  im2col(x2, nullptr, 128, 1152);
  gemm(im, 1152, 0, wp, 256, c3b, x3, 0, HWSZ, MROWS, 256, 1152, 1);

  // ---- mamba: in_proj (A read fused from NCHW) ----
  ms_pack_proj<<<blocks((long)256 * 1024), TB, 0, stream>>>(ipw, wp, 1024, 256, 256, 1024, 256L * 1024);
  gemm(x3, 0, HWSZ, wp, 1024, nullptr, xz, 1024, 0, MROWS, 1024, 256, 0);

  ms_conv1d_silu<<<blocks((long)MROWS * D_INNER), TB, 0, stream>>>(xz, c1dw, c1db, ucv,
                                                                   (long)MROWS * D_INNER);

  ms_pack_proj<<<blocks((long)512 * 144), TB, 0, stream>>>(xpw, wp, 144, 512, 512, 144, 512L * 144);
  gemm(ucv, D_INNER, 0, wp, 144, nullptr, dbl, 144, 0, MROWS, 144, 512, 0);

  ms_pack_proj<<<blocks((long)16 * 512), TB, 0, stream>>>(dtw, wp, 512, 16, 16, 512, 16L * 512);
  gemm(dbl, 144, 0, wp, 512, dtbb, dtb, 512, 0, MROWS, 512, 16, 2);  // softplus

  ms_scan<<<(BSZ * D_INNER * 32 + TB - 1) / TB, TB, 0, stream>>>(ucv, dtb, dbl, Alog, ysb);

  ms_gate<<<blocks((long)MROWS * D_INNER), TB, 0, stream>>>(ysb, ucv, xz, Dpar, yb,
                                                            (long)MROWS * D_INNER);

  ms_pack_proj<<<blocks((long)512 * 256), TB, 0, stream>>>(opw, wp, 256, 512, 512, 256, 512L * 256);
  gemm(yb, D_INNER, 0, wp, 256, nullptr, x3s, 0, HWSZ, MROWS, 256, 512, 0);

  // ---- decoder deconvs (flipped/transposed weights, residual adds fused) ----
  ms_pack_deconv<<<blocks((long)2304 * 128), TB, 0, stream>>>(d1w, wp, 128, 256, 2304, 128, 2304L * 128);
  im2col(x3s, nullptr, 256, 2304);
  gemm(im, 2304, 0, wp, 128, d1b, x4, 0, HWSZ, MROWS, 128, 2304, 1);

  ms_pack_deconv<<<blocks((long)1152 * 64), TB, 0, stream>>>(d2w, wp, 64, 128, 1152, 64, 1152L * 64);
  im2col(x4, x2, 128, 1152);
  gemm(im, 1152, 0, wp, 64, d2b, x5, 0, HWSZ, MROWS, 64, 1152, 1);

  ms_pack_deconv<<<blocks((long)576 * 16), TB, 0, stream>>>(d3w, wp, 3, 64, 576, 16, 576L * 16);
  im2col(x5, x1, 64, 576);
  gemm(im, 576, 0, wp, 16, d3b, out, 0, HWSZ, MROWS, 3, 576, 1);
}